// NeuroSAT_84370337563243
// MI455X (gfx1250) — compile-verified
//
#include <hip/hip_runtime.h>

typedef __attribute__((ext_vector_type(2))) float v2f;
typedef __attribute__((ext_vector_type(8))) float v8f;
typedef __attribute__((ext_vector_type(4))) unsigned int u32x4;
typedef __attribute__((ext_vector_type(4))) int i32x4;
typedef __attribute__((ext_vector_type(8))) int i32x8;

#define DIM 128
#define L_SZ 80000
#define C_SZ 120000
#define N_EDGES 480000
#define N_ITER 8

// GEMM flags
#define GF_RELU  1
#define GF_ACCUM 2
#define GF_FLIP  4

#define AST 68                    // A LDS row stride (64 + 4 pad) -> conflict-free b64 frags
#define WST 132                   // W^T LDS row stride (128 + 4 pad)
#define SA_CHUNK (128 * AST)      // one 128-row x 64-K A chunk (floats)
#define SMEM_FLOATS (2 * SA_CHUNK + 64 * WST)

// Tensor Data Mover: async DMA of a 2-D tile [rows x 64 floats] from global
// (row stride 128 floats) into LDS at byte offset ldsOff, inserting 4 DWORDs of
// padding after every 64 DWORDs (-> LDS row stride 68 floats). Rows beyond
// rowsValid are zero-filled by TDM OOB handling. Tracked by TENSORcnt.
__device__ __forceinline__ void tdm_load_tile(const float* gptr, unsigned ldsOff,
                                              int rowsValid)
{
    const unsigned long long ga = (unsigned long long)(uintptr_t)gptr;
    const unsigned dim1 = (unsigned)rowsValid;

    u32x4 g0;
    g0[0] = 1u;                                            // count=1, user descriptor
    g0[1] = ldsOff;                                        // lds_addr (bytes)
    g0[2] = (unsigned)(ga & 0xFFFFFFFFu);                  // global_addr[31:0]
    g0[3] = (unsigned)((ga >> 32) & 0x01FFFFFFu)           // global_addr[56:32]
          | (2u << 30);                                    // type=2 ("image")

    i32x8 g1;
    g1[0] = (int)((2u << 16)      // data_size = 4 bytes
                | (1u << 20)      // pad_enable
                | (5u << 22)      // pad_interval: 64 DWORDs
                | (3u << 25));    // pad_amount: 4 DWORDs
    g1[1] = (int)(64u << 16);                              // tensor_dim0[15:0] = 64
    g1[2] = (int)((dim1 & 0xFFFFu) << 16);                 // dim0 hi=0 | tensor_dim1[15:0]
    g1[3] = (int)((dim1 >> 16) | (64u << 16));             // tensor_dim1 hi | tile_dim0=64
    g1[4] = (int)128u;                                     // tile_dim1=128 | tile_dim2=0
    g1[5] = (int)128u;                                     // tensor_dim0_stride = 128
    g1[6] = 0;                                             // stride hi | dim1_stride lo
    g1[7] = 0;                                             // dim1_stride hi
    const i32x4 z4 = {0, 0, 0, 0};                         // groups 2/3: 2-D tensor
    const i32x8 z8 = {0, 0, 0, 0, 0, 0, 0, 0};
    __builtin_amdgcn_tensor_load_to_lds(g0, g1, z4, z4, z8, 0);
}

__device__ __forceinline__ void compute_chunk(const float* __restrict__ sA,
                                              const float* __restrict__ sW,
                                              int ktAbs, int wave, int lm, int hi,
                                              int lmA, v8f acc[4])
{
    const float* aB = sA + (wave * 16 + lmA) * AST + 2 * hi;
    const float* b0 = sW + ( 0 + lm) * WST + ktAbs + 2 * hi;
    const float* b1 = sW + (16 + lm) * WST + ktAbs + 2 * hi;
    const float* b2 = sW + (32 + lm) * WST + ktAbs + 2 * hi;
    const float* b3 = sW + (48 + lm) * WST + ktAbs + 2 * hi;
    #pragma unroll
    for (int kk = 0; kk < 64; kk += 4) {
        const v2f a = *(const v2f*)(aB + kk);
        v2f b;
        b = *(const v2f*)(b0 + kk);
        acc[0] = __builtin_amdgcn_wmma_f32_16x16x4_f32(false, a, false, b, (short)0, acc[0], false, false);
        b = *(const v2f*)(b1 + kk);
        acc[1] = __builtin_amdgcn_wmma_f32_16x16x4_f32(false, a, false, b, (short)0, acc[1], false, false);
        b = *(const v2f*)(b2 + kk);
        acc[2] = __builtin_amdgcn_wmma_f32_16x16x4_f32(false, a, false, b, (short)0, acc[2], false, false);
        b = *(const v2f*)(b3 + kk);
        acc[3] = __builtin_amdgcn_wmma_f32_16x16x4_f32(false, a, false, b, (short)0, acc[3], false, false);
    }
}

// C[M,N] (+)= op(A[M,128] @ W[128,N] + bias), W row-major.
// flags: GF_RELU, GF_ACCUM (C += result), GF_FLIP (A row index ^= 1, applied at LDS read)
__global__ __launch_bounds__(256) void gemm_k128(
    const float* __restrict__ A, const float* __restrict__ W,
    const float* __restrict__ bias, float* __restrict__ C,
    int M, int N, int flags)
{
    extern __shared__ float smem[];
    float* sA0 = smem;                     // [128][AST]  K chunk 0
    float* sA1 = smem + SA_CHUNK;          // [128][AST]  K chunk 1
    float* sW  = smem + 2 * SA_CHUNK;      // [64][WST]   W^T block [n][k], full K

    const int tid  = threadIdx.x;
    const int lane = tid & 31;
    const int wave = tid >> 5;             // 0..7
    const int lm   = lane & 15;
    const int hi   = lane >> 4;            // 0/1
    const int R0   = blockIdx.x * 128;
    const int N0   = blockIdx.y * 64;
    const int lmA  = (flags & GF_FLIP) ? (lm ^ 1) : lm;   // l2l pos/neg literal swap

    // ---- issue async TDM loads for both A chunks (wave 0 only)
    if (tid < 32) {
        tdm_load_tile(A + (size_t)R0 * DIM,      0u,                       M - R0);
        tdm_load_tile(A + (size_t)R0 * DIM + 64, (unsigned)(SA_CHUNK * 4), M - R0);
        __builtin_amdgcn_s_wait_tensorcnt(1);   // chunk 0 resident; chunk 1 in flight
    }

    // ---- stage W^T block while TDM runs: n in [N0,N0+64), k in [0,128)
    {
        const int n  = tid & 63;
        const int k0 = tid >> 6;           // 0..3
        const float* wp = W + (size_t)k0 * N + N0 + n;
        #pragma unroll
        for (int p = 0; p < 32; ++p)
            sW[n * WST + k0 + p * 4] = wp[(size_t)p * 4 * N];
    }
    __syncthreads();

    v8f acc[4];
    acc[0] = v8f{}; acc[1] = v8f{}; acc[2] = v8f{}; acc[3] = v8f{};

    compute_chunk(sA0, sW, 0, wave, lm, hi, lmA, acc);       // overlaps chunk-1 TDM
    if (tid < 32) __builtin_amdgcn_s_wait_tensorcnt(0);
    __syncthreads();
    compute_chunk(sA1, sW, 64, wave, lm, hi, lmA, acc);

    // epilogue: C/D layout: VGPR r -> M = r + 8*hi (within 16-row wave tile), N = lm
    const int rowBase = R0 + wave * 16 + 8 * hi;
    #pragma unroll
    for (int j = 0; j < 4; ++j) {
        const int col = N0 + j * 16 + lm;
        const float bv = bias ? bias[col] : 0.0f;
        #pragma unroll
        for (int r = 0; r < 8; ++r) {
            const int row = rowBase + r;
            if (row < M) {
                float v = acc[j][r] + bv;
                if (flags & GF_RELU) v = fmaxf(v, 0.0f);
                const size_t off = (size_t)row * N + col;
                if (flags & GF_ACCUM) v += C[off];
                C[off] = v;
            }
        }
    }
}

// aggr[dst[e]] += msg[src[e]]  (1 thread per edge x float4 chunk)
__global__ __launch_bounds__(256) void scatter_add(
    const float* __restrict__ msg, const int* __restrict__ src_idx,
    const int* __restrict__ dst_idx, float* __restrict__ aggr, int nEdges)
{
    const int t = blockIdx.x * blockDim.x + threadIdx.x;
    const int e = t >> 5;
    if (e >= nEdges) return;
    const int c = (t & 31) * 4;
    const float4 v = *(const float4*)(msg + (size_t)src_idx[e] * DIM + c);
    float* d = aggr + (size_t)dst_idx[e] * DIM + c;
    unsafeAtomicAdd(d + 0, v.x);
    unsafeAtomicAdd(d + 1, v.y);
    unsafeAtomicAdd(d + 2, v.z);
    unsafeAtomicAdd(d + 3, v.w);
}

__device__ __forceinline__ float sigm(float x) { return 1.0f / (1.0f + __expf(-x)); }

// LayerNorm-LSTM cell pointwise: one 128-thread block per row.
__global__ __launch_bounds__(128) void lstm_point(
    const float* __restrict__ gx, const float* __restrict__ gh,
    const float* __restrict__ g_ih, const float* __restrict__ b_ih,
    const float* __restrict__ g_hh, const float* __restrict__ b_hh,
    const float* __restrict__ g_c,  const float* __restrict__ b_c,
    float* __restrict__ cstate, float* __restrict__ hout, int M)
{
    __shared__ float r0[4], r1[4], r2[4], r3[4];
    const int row = blockIdx.x;
    if (row >= M) return;
    const int p = threadIdx.x;           // 0..127
    const size_t gbase = (size_t)row * 512;

    float xv[4], hv[4];
    float sx = 0.f, sxx = 0.f, sh = 0.f, shh = 0.f;
    #pragma unroll
    for (int k = 0; k < 4; ++k) {
        xv[k] = gx[gbase + p + 128 * k];
        hv[k] = gh[gbase + p + 128 * k];
        sx += xv[k]; sxx += xv[k] * xv[k];
        sh += hv[k]; shh += hv[k] * hv[k];
    }
    #pragma unroll
    for (int off = 16; off; off >>= 1) {
        sx  += __shfl_xor(sx,  off); sxx += __shfl_xor(sxx, off);
        sh  += __shfl_xor(sh,  off); shh += __shfl_xor(shh, off);
    }
    const int wid = p >> 5;
    if ((p & 31) == 0) { r0[wid] = sx; r1[wid] = sxx; r2[wid] = sh; r3[wid] = shh; }
    __syncthreads();
    sx  = r0[0] + r0[1] + r0[2] + r0[3];
    sxx = r1[0] + r1[1] + r1[2] + r1[3];
    sh  = r2[0] + r2[1] + r2[2] + r2[3];
    shh = r3[0] + r3[1] + r3[2] + r3[3];
    __syncthreads();

    const float mux = sx * (1.0f / 512.0f);
    const float muh = sh * (1.0f / 512.0f);
    const float rx = rsqrtf(sxx * (1.0f / 512.0f) - mux * mux + 1e-5f);
    const float rh = rsqrtf(shh * (1.0f / 512.0f) - muh * muh + 1e-5f);

    float gate[4];
    #pragma unroll
    for (int k = 0; k < 4; ++k) {
        const int idx = p + 128 * k;
        gate[k] = (xv[k] - mux) * rx * g_ih[idx] + b_ih[idx]
                + (hv[k] - muh) * rh * g_hh[idx] + b_hh[idx];
    }
    const float gi = gate[0], gf = gate[1], gg = gate[2], go = gate[3];
    const size_t cbase = (size_t)row * DIM + p;
    const float cold = cstate[cbase];
    const float nc = sigm(gf) * cold + sigm(gi) * tanhf(gg);
    cstate[cbase] = nc;

    float s = nc, ss = nc * nc;
    #pragma unroll
    for (int off = 16; off; off >>= 1) {
        s += __shfl_xor(s, off); ss += __shfl_xor(ss, off);
    }
    if ((p & 31) == 0) { r0[wid] = s; r1[wid] = ss; }
    __syncthreads();
    s  = r0[0] + r0[1] + r0[2] + r0[3];
    ss = r1[0] + r1[1] + r1[2] + r1[3];
    const float mu = s * (1.0f / 128.0f);
    const float rc = rsqrtf(ss * (1.0f / 128.0f) - mu * mu + 1e-5f);
    hout[cbase] = sigm(go) * tanhf((nc - mu) * rc * g_c[p] + b_c[p]);
}

extern "C" void kernel_launch(void* const* d_in, const int* in_sizes, int n_in,
                              void* d_out, int out_size, void* d_ws, size_t ws_size,
                              hipStream_t stream) {
    (void)in_sizes; (void)n_in; (void)out_size; (void)ws_size;

    const int* l_edge = (const int*)d_in[2];
    const int* c_edge = (const int*)d_in[3];
    const float* l_emb0 = (const float*)d_in[4];
    const float* c_emb0 = (const float*)d_in[5];
    const float* l2c_w1 = (const float*)d_in[6];
    const float* l2c_b1 = (const float*)d_in[7];
    const float* l2c_w2 = (const float*)d_in[8];
    const float* l2c_b2 = (const float*)d_in[9];
    const float* c2l_w1 = (const float*)d_in[10];
    const float* c2l_b1 = (const float*)d_in[11];
    const float* c2l_w2 = (const float*)d_in[12];
    const float* c2l_b2 = (const float*)d_in[13];
    const float* cu_wi  = (const float*)d_in[14];
    const float* cu_wh  = (const float*)d_in[15];
    const float* cu_g_ih = (const float*)d_in[16];
    const float* cu_b_ih = (const float*)d_in[17];
    const float* cu_g_hh = (const float*)d_in[18];
    const float* cu_b_hh = (const float*)d_in[19];
    const float* cu_g_c  = (const float*)d_in[20];
    const float* cu_b_c  = (const float*)d_in[21];
    const float* lu_wi   = (const float*)d_in[22];   // [256,512] row-major
    const float* lu_wh   = (const float*)d_in[23];
    const float* lu_g_ih = (const float*)d_in[24];
    const float* lu_b_ih = (const float*)d_in[25];
    const float* lu_g_hh = (const float*)d_in[26];
    const float* lu_b_hh = (const float*)d_in[27];
    const float* lu_g_c  = (const float*)d_in[28];
    const float* lu_b_c  = (const float*)d_in[29];

    float* out = (float*)d_out;
    const size_t LD = (size_t)L_SZ * DIM;
    const size_t CD = (size_t)C_SZ * DIM;

    // workspace layout
    float* ws = (float*)d_ws;
    float* c_state = ws;                        // CD
    float* l_state = c_state + CD;              // LD
    float* aggr    = l_state + LD;              // CD (shared c/l aggregation)
    float* bufA    = aggr + CD;                 // C_SZ*512 (MLP tmp / gates_x)
    float* bufB    = bufA + (size_t)C_SZ * 512; // C_SZ*512 (msg / gates_h)

    const int smem = SMEM_FLOATS * sizeof(float);   // ~101 KB (WGP has 320 KB)
    (void)hipFuncSetAttribute((const void*)gemm_k128,
                              hipFuncAttributeMaxDynamicSharedMemorySize, smem);
    auto gemm = [&](const float* A, const float* W, const float* bias,
                    float* Cp, int M, int N, int flags) {
        dim3 grid((M + 127) / 128, N / 64);
        gemm_k128<<<grid, 256, smem, stream>>>(A, W, bias, Cp, M, N, flags);
    };
    const int scatBlocks = (N_EDGES * 32 + 255) / 256;

    // init: zero LSTM states, copy input embeddings into output slot 0
    (void)hipMemsetAsync(c_state, 0, CD * sizeof(float), stream);
    (void)hipMemsetAsync(l_state, 0, LD * sizeof(float), stream);
    (void)hipMemcpyAsync(out,          l_emb0, LD * sizeof(float), hipMemcpyDeviceToDevice, stream);
    (void)hipMemcpyAsync(out + 9 * LD, c_emb0, CD * sizeof(float), hipMemcpyDeviceToDevice, stream);

    for (int it = 0; it < N_ITER; ++it) {
        const float* l_prev = out + (size_t)it * LD;
        const float* c_prev = out + 9 * LD + (size_t)it * CD;
        float* l_next = out + (size_t)(it + 1) * LD;
        float* c_next = out + 9 * LD + (size_t)(it + 1) * CD;

        // ---- literal -> clause
        gemm(l_prev, l2c_w1, l2c_b1, bufA, L_SZ, DIM, GF_RELU);      // hidden
        gemm(bufA,   l2c_w2, l2c_b2, bufB, L_SZ, DIM, 0);            // l_msg_feat
        (void)hipMemsetAsync(aggr, 0, CD * sizeof(float), stream);
        scatter_add<<<scatBlocks, 256, 0, stream>>>(bufB, l_edge, c_edge, aggr, N_EDGES);

        // clause LN-LSTM
        gemm(aggr,   cu_wi, nullptr, bufA, C_SZ, 512, 0);            // gates_x
        gemm(c_prev, cu_wh, nullptr, bufB, C_SZ, 512, 0);            // gates_h
        lstm_point<<<C_SZ, 128, 0, stream>>>(bufA, bufB,
            cu_g_ih, cu_b_ih, cu_g_hh, cu_b_hh, cu_g_c, cu_b_c,
            c_state, c_next, C_SZ);

        // ---- clause -> literal
        gemm(c_next, c2l_w1, c2l_b1, bufA, C_SZ, DIM, GF_RELU);
        gemm(bufA,   c2l_w2, c2l_b2, bufB, C_SZ, DIM, 0);            // c_msg_feat
        (void)hipMemsetAsync(aggr, 0, LD * sizeof(float), stream);
        scatter_add<<<scatBlocks, 256, 0, stream>>>(bufB, c_edge, l_edge, aggr, N_EDGES);

        // literal LN-LSTM: x = [c2l_aggr | l2l_msg]; l2l_msg[r] = l_prev[r^1]
        gemm(aggr,   lu_wi,                     nullptr, bufA, L_SZ, 512, 0);
        gemm(l_prev, lu_wi + (size_t)128 * 512, nullptr, bufA, L_SZ, 512, GF_ACCUM | GF_FLIP);
        gemm(l_prev, lu_wh,                     nullptr, bufB, L_SZ, 512, 0);
        lstm_point<<<L_SZ, 128, 0, stream>>>(bufA, bufB,
            lu_g_ih, lu_b_ih, lu_g_hh, lu_b_hh, lu_g_c, lu_b_c,
            l_state, l_next, L_SZ);
    }
}